// HiddenRRGCN_52218212384772
// MI455X (gfx1250) — compile-verified
//
#include <hip/hip_runtime.h>

#define DD 128
#define SLOPE 0.05f
#define LN_EPS 1e-5f

typedef float v2f __attribute__((ext_vector_type(2)));
typedef float v8f __attribute__((ext_vector_type(8)));

// ---------------------------------------------------------------- zero fill
__global__ __launch_bounds__(256) void zero_f4(float4* __restrict__ p, size_t n4) {
    size_t i = (size_t)blockIdx.x * blockDim.x + threadIdx.x;
    size_t stride = (size_t)gridDim.x * blockDim.x;
    float4 z = make_float4(0.f, 0.f, 0.f, 0.f);
    for (; i < n4; i += stride) p[i] = z;
}

// ---------------------------------------------------------------- dual GEMM
// O1 = A @ W1, O2 = A @ W2.  A:[nrows,128], W:[128,128] row-major.
// Block = 8 waves; block owns 16 rows, wave w owns column tile w (16 cols).
// Uses V_WMMA_F32_16X16X4_F32 (exact fp32), K stepped by 4, 32 steps.
__global__ __launch_bounds__(256) void gemm_dual(
    const float* __restrict__ A,
    const float* __restrict__ W1, const float* __restrict__ W2,
    float* __restrict__ O1, float* __restrict__ O2, int nrows) {
    const int wave = threadIdx.x >> 5;        // 0..7 -> column tile
    const int lane = threadIdx.x & 31;
    const int half = lane >> 4;               // 0 or 1
    const int l15  = lane & 15;
    const int rowBase = blockIdx.x * 16;
    if (rowBase >= nrows) return;             // uniform per block

    const int rowLd = min(rowBase + l15, nrows - 1);   // clamped load row (M = l15)
    const int col   = wave * 16 + l15;                 // N = l15
    const float* __restrict__ arow = A + (size_t)rowLd * DD;

    v8f acc1 = {};
    v8f acc2 = {};
    #pragma unroll 8
    for (int k0 = 0; k0 < DD; k0 += 4) {
        const int kb = k0 + (half << 1);               // K pair base for this lane
        v2f a = *(const v2f*)(arow + kb);              // A[M, kb], A[M, kb+1]
        v2f b1, b2;
        b1.x = W1[(size_t)kb * DD + col];
        b1.y = W1[(size_t)(kb + 1) * DD + col];
        b2.x = W2[(size_t)kb * DD + col];
        b2.y = W2[(size_t)(kb + 1) * DD + col];
        acc1 = __builtin_amdgcn_wmma_f32_16x16x4_f32(false, a, false, b1,
                                                     (short)0, acc1, false, false);
        acc2 = __builtin_amdgcn_wmma_f32_16x16x4_f32(false, a, false, b2,
                                                     (short)0, acc2, false, false);
    }
    // C/D layout: VGPR i -> row (i + 8*half), col = l15 within the tile
    #pragma unroll
    for (int i = 0; i < 8; ++i) {
        const int r = rowBase + i + half * 8;
        if (r < nrows) {
            O1[(size_t)r * DD + col] = acc1[i];
            O2[(size_t)r * DD + col] = acc2[i];
        }
    }
}

// ---------------------------------------------------------------- edge scatter
// One wave per edge: gather X[src]*, atomically add into agg[dst]*, count degree.
__global__ __launch_bounds__(256) void scatter_edges(
    const float* __restrict__ X, const int* __restrict__ src,
    const int* __restrict__ dst, float* __restrict__ agg,
    float* __restrict__ deg, int nE) {
    const int e = blockIdx.x * 8 + (threadIdx.x >> 5);
    const int lane = threadIdx.x & 31;
    if (e >= nE) return;
    const int s = src[e];
    const int d = dst[e];
    const float4 v = *(const float4*)(X + (size_t)s * DD + lane * 4);
    float* out = agg + (size_t)d * DD + lane * 4;
    atomicAdd(out + 0, v.x);
    atomicAdd(out + 1, v.y);
    atomicAdd(out + 2, v.z);
    atomicAdd(out + 3, v.w);
    if (lane == 0) atomicAdd(deg + d, 1.0f);
}

// ---------------------------------------------------------------- finalize
// out = LN(leaky(agg1/max(deg1,1) + agg2/max(deg2,1) + bias)) * gamma + beta
// One wave per node, 4 channels per lane, wave32 butterfly reduction.
__global__ __launch_bounds__(256) void finalize_nodes(
    const float* __restrict__ agg1, const float* __restrict__ deg1,
    const float* __restrict__ agg2, const float* __restrict__ deg2,
    const float* __restrict__ bias, const float* __restrict__ gamma,
    const float* __restrict__ beta, float* __restrict__ out, int n) {
    const int node = blockIdx.x * 8 + (threadIdx.x >> 5);
    const int lane = threadIdx.x & 31;
    if (node >= n) return;
    const float r1 = 1.0f / fmaxf(deg1[node], 1.0f);
    const float r2 = 1.0f / fmaxf(deg2[node], 1.0f);
    const size_t base = (size_t)node * DD + lane * 4;
    const float4 a  = *(const float4*)(agg1 + base);
    const float4 b  = *(const float4*)(agg2 + base);
    const float4 bb = *(const float4*)(bias + lane * 4);
    float x[4];
    x[0] = a.x * r1 + b.x * r2 + bb.x;
    x[1] = a.y * r1 + b.y * r2 + bb.y;
    x[2] = a.z * r1 + b.z * r2 + bb.z;
    x[3] = a.w * r1 + b.w * r2 + bb.w;
    float s = 0.f, q = 0.f;
    #pragma unroll
    for (int i = 0; i < 4; ++i) {
        x[i] = (x[i] >= 0.f) ? x[i] : SLOPE * x[i];   // LeakyReLU
        s += x[i];
        q += x[i] * x[i];
    }
    #pragma unroll
    for (int m = 16; m > 0; m >>= 1) {
        s += __shfl_xor(s, m, 32);
        q += __shfl_xor(q, m, 32);
    }
    const float mu   = s * (1.0f / DD);
    const float var  = q * (1.0f / DD) - mu * mu;
    const float rstd = rsqrtf(var + LN_EPS);
    const float4 g  = *(const float4*)(gamma + lane * 4);
    const float4 be = *(const float4*)(beta + lane * 4);
    float4 y;
    y.x = (x[0] - mu) * rstd * g.x + be.x;
    y.y = (x[1] - mu) * rstd * g.y + be.y;
    y.z = (x[2] - mu) * rstd * g.z + be.z;
    y.w = (x[3] - mu) * rstd * g.w + be.w;
    *(float4*)(out + base) = y;
}

// ---------------------------------------------------------------- launch
extern "C" void kernel_launch(void* const* d_in, const int* in_sizes, int n_in,
                              void* d_out, int out_size, void* d_ws, size_t ws_size,
                              hipStream_t stream) {
    const float* h_cell = (const float*)d_in[0];
    const float* h_gene = (const float*)d_in[1];
    const int* src_cg = (const int*)d_in[2];
    const int* dst_cg = (const int*)d_in[3];
    const int* src_gc = (const int*)d_in[4];
    const int* dst_gc = (const int*)d_in[5];
    const int* src_cc = (const int*)d_in[6];
    const int* dst_cc = (const int*)d_in[7];
    const int* src_gg = (const int*)d_in[8];
    const int* dst_gg = (const int*)d_in[9];
    const float* W_cg = (const float*)d_in[10];
    const float* W_gc = (const float*)d_in[11];
    const float* W_cc = (const float*)d_in[12];
    const float* W_gg = (const float*)d_in[13];
    const float* b_cell    = (const float*)d_in[14];
    const float* b_gene    = (const float*)d_in[15];
    const float* ln_s_cell = (const float*)d_in[16];
    const float* ln_b_cell = (const float*)d_in[17];
    const float* ln_s_gene = (const float*)d_in[18];
    const float* ln_b_gene = (const float*)d_in[19];

    const int NC = in_sizes[0] / DD;
    const int NG = in_sizes[1] / DD;
    const int E_CG = in_sizes[2];
    const int E_GC = in_sizes[4];
    const int E_CC = in_sizes[6];
    const int E_GG = in_sizes[8];

    // Workspace layout (floats). Zeroed region (agg + deg) is contiguous first.
    float* ws = (float*)d_ws;
    size_t off = 0;
    float* agg_c1 = ws + off; off += (size_t)NC * DD;   // from gc edges
    float* agg_c2 = ws + off; off += (size_t)NC * DD;   // from cc edges
    float* agg_g1 = ws + off; off += (size_t)NG * DD;   // from cg edges
    float* agg_g2 = ws + off; off += (size_t)NG * DD;   // from gg edges
    float* deg_c1 = ws + off; off += (size_t)NC;
    float* deg_c2 = ws + off; off += (size_t)NC;
    float* deg_g1 = ws + off; off += (size_t)NG;
    float* deg_g2 = ws + off; off += (size_t)NG;
    const size_t zeroFloats = off;
    float* Xc_cc = ws + off; off += (size_t)NC * DD;    // hc @ W_cc
    float* Xc_cg = ws + off; off += (size_t)NC * DD;    // hc @ W_cg
    float* Xg_gc = ws + off; off += (size_t)NG * DD;    // hg @ W_gc
    float* Xg_gg = ws + off; off += (size_t)NG * DD;    // hg @ W_gg

    float* out_c = (float*)d_out;                        // [NC, D]
    float* out_g = (float*)d_out + (size_t)NC * DD;      // [NG, D]

    const float* hc = h_cell;
    const float* hg = h_gene;

    const dim3 blk(256);
    const int gemmBlocksC = (NC + 15) / 16;
    const int gemmBlocksG = (NG + 15) / 16;

    for (int layer = 0; layer < 2; ++layer) {
        zero_f4<<<2048, blk, 0, stream>>>((float4*)ws, zeroFloats / 4);

        gemm_dual<<<gemmBlocksC, blk, 0, stream>>>(hc, W_cc, W_cg, Xc_cc, Xc_cg, NC);
        gemm_dual<<<gemmBlocksG, blk, 0, stream>>>(hg, W_gc, W_gg, Xg_gc, Xg_gg, NG);

        scatter_edges<<<(E_GC + 7) / 8, blk, 0, stream>>>(Xg_gc, src_gc, dst_gc, agg_c1, deg_c1, E_GC);
        scatter_edges<<<(E_CC + 7) / 8, blk, 0, stream>>>(Xc_cc, src_cc, dst_cc, agg_c2, deg_c2, E_CC);
        scatter_edges<<<(E_CG + 7) / 8, blk, 0, stream>>>(Xc_cg, src_cg, dst_cg, agg_g1, deg_g1, E_CG);
        scatter_edges<<<(E_GG + 7) / 8, blk, 0, stream>>>(Xg_gg, src_gg, dst_gg, agg_g2, deg_g2, E_GG);

        finalize_nodes<<<(NC + 7) / 8, blk, 0, stream>>>(agg_c1, deg_c1, agg_c2, deg_c2,
                                                         b_cell, ln_s_cell, ln_b_cell, out_c, NC);
        finalize_nodes<<<(NG + 7) / 8, blk, 0, stream>>>(agg_g1, deg_g1, agg_g2, deg_g2,
                                                         b_gene, ln_s_gene, ln_b_gene, out_g, NG);

        hc = out_c;   // layer-2 reads layer-1 output from d_out, then overwrites it
        hg = out_g;
    }
}